// AttentionGate_86449101734209
// MI455X (gfx1250) — compile-verified
//
#include <hip/hip_runtime.h>

// ---------------------------------------------------------------------------
// AttentionGate for MI455X (gfx1250, wave32).
//   Stage 1: q,k,v projection GEMMs via V_WMMA_F32_16X16X4_F32 into ws.
//   Stage 2: fused flash attention, double-buffered K/V tiles staged in LDS
//            via GLOBAL_LOAD_ASYNC_TO_LDS_* (ASYNCcnt), energy + PV on WMMA,
//            online softmax in fp32.
// ---------------------------------------------------------------------------

typedef __attribute__((ext_vector_type(2))) float v2f;
typedef __attribute__((ext_vector_type(8))) float v8f;

constexpr int BATCH = 4;
constexpr int C     = 256;
constexpr int C8    = 32;
constexpr int HW    = 4096;

constexpr int KB    = 16;    // keys per pipelined tile
constexpr int KSTR2 = 20;    // K-tile LDS row stride (dwords): 16B-aligned rows,
constexpr int VSTR  = 20;    // gcd(20,64)=4 -> conflict-free column reads
constexpr int SSTR  = 17;    // S-transpose scratch stride

__device__ __forceinline__ v8f wmma_f32(v2f a, v2f b, v8f c) {
  return __builtin_amdgcn_wmma_f32_16x16x4_f32(false, a, false, b, (short)0, c,
                                               false, false);
}

// --- gfx1250 async global->LDS (bypasses VGPRs, tracked by ASYNCcnt) -------
__device__ __forceinline__ void async_ld_b128(unsigned lds_off, const float* g) {
  asm volatile("global_load_async_to_lds_b128 %0, %1, off"
               :: "v"(lds_off), "v"(g) : "memory");
}
__device__ __forceinline__ void async_ld_b64(unsigned lds_off, const float* g) {
  asm volatile("global_load_async_to_lds_b64 %0, %1, off"
               :: "v"(lds_off), "v"(g) : "memory");
}
__device__ __forceinline__ void wait_async_le5() {
  asm volatile("s_wait_asynccnt 0x5" ::: "memory");
}
__device__ __forceinline__ void wait_async_0() {
  asm volatile("s_wait_asynccnt 0x0" ::: "memory");
}
__device__ __forceinline__ unsigned lds_off(const float* p) {
  return (unsigned)(uintptr_t)p;  // generic shared ptr: low 32 bits = LDS offset
}

// ---------------------------------------------------------------------------
// Projection: Y[b][m][n] = sum_c W[m][c] * X[b][c][n] + bias[m]
// grid = (HW/128, M/16, BATCH), block = 256 (8 waves, one 16x16 tile each).
// ---------------------------------------------------------------------------
__global__ __launch_bounds__(256) void ag_proj_kernel(
    const float* __restrict__ Wm, const float* __restrict__ bias,
    const float* __restrict__ X, float* __restrict__ Y, int M) {
  const int lane  = threadIdx.x & 31;
  const int wave  = threadIdx.x >> 5;
  const int lhalf = lane >> 4;
  const int lmod  = lane & 15;
  const int b     = blockIdx.z;
  const int m0    = blockIdx.y * 16;
  const int n0    = (blockIdx.x * 8 + wave) * 16;

  const float* Xb = X + (size_t)b * C * HW;
  float*       Yb = Y + (size_t)b * M * HW;

  const float* wrow = Wm + (size_t)(m0 + lmod) * C;
  const float* xcol = Xb + n0 + lmod;

  v8f acc = {};
#pragma unroll 8
  for (int k0 = 0; k0 < C; k0 += 4) {
    const int ka = k0 + 2 * lhalf;
    v2f a, bb;
    a.x  = wrow[ka];
    a.y  = wrow[ka + 1];
    bb.x = xcol[(size_t)ka * HW];
    bb.y = xcol[(size_t)(ka + 1) * HW];
    acc = wmma_f32(a, bb, acc);
  }
#pragma unroll
  for (int r = 0; r < 8; ++r) {
    const int mr = m0 + r + 8 * lhalf;
    Yb[(size_t)mr * HW + n0 + lmod] = acc[r] + bias[mr];
  }
}

// ---------------------------------------------------------------------------
// Fused flash attention. grid = (HW/128, BATCH), block = 256 (8 waves).
// Double-buffered 16-key K/V tiles staged in LDS by the async engine.
// ---------------------------------------------------------------------------
__global__ __launch_bounds__(256) void ag_attn_kernel(
    const float* __restrict__ qws, const float* __restrict__ kws,
    const float* __restrict__ vws, const float* __restrict__ x,
    const float* __restrict__ gamma, float* __restrict__ out) {
  __shared__ float kbuf[2][C8 * KSTR2];   //  2 x 2.5 KB
  __shared__ float vbuf[2][C * VSTR];     //  2 x 20  KB
  __shared__ float slds[8 * 16 * SSTR];   //  8.5 KB S-transpose scratch

  const int lane  = threadIdx.x & 31;
  const int wave  = threadIdx.x >> 5;
  const int lhalf = lane >> 4;
  const int lmod  = lane & 15;
  const int b     = blockIdx.y;
  const int q0    = blockIdx.x * 128 + wave * 16;

  const float* qp = qws + (size_t)b * C8 * HW;
  const float* kp = kws + (size_t)b * C8 * HW;
  const float* vp = vws + (size_t)b * C  * HW;

  // --- issue one tile's worth of async loads: 5 instructions per thread ---
  auto issue_tile = [&](int buf, int kb) {
    const int o = threadIdx.x;
#pragma unroll
    for (int u = 0; u < 4; ++u) {             // V: 256 rows x 16 keys, b128
      const int idx = o + 256 * u;            // 1024 16-byte chunks
      const int c   = idx >> 2;
      const int j   = (idx & 3) * 4;
      async_ld_b128(lds_off(&vbuf[buf][c * VSTR + j]),
                    vp + (size_t)c * HW + kb + j);
    }
    {                                          // K: 32 rows x 16 keys, b64
      const int c = o >> 3;
      const int j = (o & 7) * 2;
      async_ld_b64(lds_off(&kbuf[buf][c * KSTR2 + j]),
                   kp + (size_t)c * HW + kb + j);
    }
  };

  // Q fragments: A of energy-WMMA, K-dim = C8 = 32 -> 8 steps of 4.
  v2f afrag[8];
#pragma unroll
  for (int kc = 0; kc < 8; ++kc) {
    const int ca = 4 * kc + 2 * lhalf;
    afrag[kc].x = qp[(size_t)ca * HW + q0 + lmod];
    afrag[kc].y = qp[(size_t)(ca + 1) * HW + q0 + lmod];
  }

  v8f oacc[16] = {};        // O tile: 256 channels x 16 queries
  float mrow = -3.0e38f;
  float lrow = 0.0f;
  float* smy = &slds[wave * 16 * SSTR];

  constexpr int NT = HW / KB;   // 256 pipelined tiles
  issue_tile(0, 0);             // prologue

#pragma unroll 1
  for (int it = 0; it < NT; ++it) {
    const int cur = it & 1;
    if (it + 1 < NT) {
      issue_tile(cur ^ 1, (it + 1) * KB);   // prefetch next tile
      wait_async_le5();                     // tile `it` is now in LDS
    } else {
      wait_async_0();
    }
    __syncthreads();                        // all waves' async writes visible

    const float* kt = kbuf[cur];
    const float* vt = vbuf[cur];

    // ---- S = Q^T K : 16 queries x 16 keys ----
    v8f s = {};
#pragma unroll
    for (int kc = 0; kc < 8; ++kc) {
      const int ca = 4 * kc + 2 * lhalf;
      v2f bb;
      bb.x = kt[ca * KSTR2 + lmod];
      bb.y = kt[(ca + 1) * KSTR2 + lmod];
      s = wmma_f32(afrag[kc], bb, s);
    }

    // ---- transpose S via LDS: D layout (key=lane) -> [query][key] ----
#pragma unroll
    for (int r = 0; r < 8; ++r)
      smy[(r + 8 * lhalf) * SSTR + lmod] = s[r];
    asm volatile("s_wait_dscnt 0" ::: "memory");  // same-wave LDS is in-order

    float pv[8];
    float tmax = -3.0e38f;
#pragma unroll
    for (int kc2 = 0; kc2 < 4; ++kc2)
#pragma unroll
      for (int h = 0; h < 2; ++h) {
        const float sv = smy[lmod * SSTR + 4 * kc2 + 2 * lhalf + h];
        pv[kc2 * 2 + h] = sv;
        tmax = fmaxf(tmax, sv);
      }
    tmax = fmaxf(tmax, __shfl_xor(tmax, 16, 32));  // lane pair shares a query

    // ---- online softmax update ----
    const float mnew = fmaxf(mrow, tmax);
    const float corr = __expf(mrow - mnew);
    float psum = 0.0f;
#pragma unroll
    for (int j = 0; j < 8; ++j) {
      pv[j] = __expf(pv[j] - mnew);
      psum += pv[j];
    }
    psum += __shfl_xor(psum, 16, 32);
    lrow = lrow * corr + psum;
    mrow = mnew;
#pragma unroll
    for (int ct = 0; ct < 16; ++ct)
#pragma unroll
      for (int r = 0; r < 8; ++r) oacc[ct][r] *= corr;

    // ---- O += V * P^T : V fragments from LDS (ds_load_b64) ----
#pragma unroll
    for (int kc2 = 0; kc2 < 4; ++kc2) {
      v2f bb;
      bb.x = pv[kc2 * 2 + 0];
      bb.y = pv[kc2 * 2 + 1];
      const int jcol = 4 * kc2 + 2 * lhalf;
#pragma unroll
      for (int ct = 0; ct < 16; ++ct) {
        const v2f av = *(const v2f*)&vt[(16 * ct + lmod) * VSTR + jcol];
        oacc[ct] = wmma_f32(av, bb, oacc[ct]);
      }
    }

    __syncthreads();   // done reading `cur`; next iter may overwrite it
  }

  // ---- finalize: out = gamma * O / l + x ----
  const float gm   = gamma[0];
  const float rinv = 1.0f / lrow;
  const float* xb  = x   + (size_t)b * C * HW;
  float*       ob  = out + (size_t)b * C * HW;
#pragma unroll
  for (int ct = 0; ct < 16; ++ct)
#pragma unroll
    for (int r = 0; r < 8; ++r) {
      const int c = 16 * ct + r + 8 * lhalf;
      const size_t idx = (size_t)c * HW + q0 + lmod;
      ob[idx] = gm * (oacc[ct][r] * rinv) + xb[idx];
    }
}

// ---------------------------------------------------------------------------
extern "C" void kernel_launch(void* const* d_in, const int* in_sizes, int n_in,
                              void* d_out, int out_size, void* d_ws,
                              size_t ws_size, hipStream_t stream) {
  const float* x     = (const float*)d_in[0];
  const float* g     = (const float*)d_in[1];
  const float* q_w   = (const float*)d_in[2];
  const float* q_b   = (const float*)d_in[3];
  const float* k_w   = (const float*)d_in[4];
  const float* k_b   = (const float*)d_in[5];
  const float* v_w   = (const float*)d_in[6];
  const float* v_b   = (const float*)d_in[7];
  const float* gamma = (const float*)d_in[8];
  float* out = (float*)d_out;

  // Workspace: q (2MB) | k (2MB) | v (16MB) — L2-resident on MI455X.
  float* qws = (float*)d_ws;
  float* kws = qws + (size_t)BATCH * C8 * HW;
  float* vws = kws + (size_t)BATCH * C8 * HW;

  dim3 blk(256);
  ag_proj_kernel<<<dim3(HW / 128, C8 / 16, BATCH), blk, 0, stream>>>(
      q_w, q_b, x, qws, C8);
  ag_proj_kernel<<<dim3(HW / 128, C8 / 16, BATCH), blk, 0, stream>>>(
      k_w, k_b, g, kws, C8);
  ag_proj_kernel<<<dim3(HW / 128, C / 16, BATCH), blk, 0, stream>>>(
      v_w, v_b, g, vws, C);
  ag_attn_kernel<<<dim3(HW / 128, BATCH), blk, 0, stream>>>(
      qws, kws, vws, x, gamma, out);
}